// Attention_661424964229
// MI455X (gfx1250) — compile-verified
//
#include <hip/hip_runtime.h>

typedef __attribute__((ext_vector_type(2))) float v2f;
typedef __attribute__((ext_vector_type(4))) float v4f;
typedef __attribute__((ext_vector_type(8))) float v8f;

#define ROW_LEN 4096  // elements per batch row

// out[b,i] = x[b,i] * (sum_k x[b,k]^2) / 16
// One block per batch row (8 blocks), 256 threads = 8 wave32.
// Sum-of-squares reduction runs through V_WMMA_F32_16X16X4_F32:
//   A = squared elements (sum is permutation invariant, so the documented
//       16x4 f32 A layout slot assignment is irrelevant as long as every
//       element lands in exactly one slot),
//   B = all ones  => D[i][j] = sum_k A[i][k] for every column j (layout-proof),
//   C accumulates across the row chunks.
// A lane's 8 C registers hold one half-column of D (documented D layout);
// lane L + lane L^16 complete the column => full partial sum in every lane.
__global__ __launch_bounds__(256) void sumsq_scale_wmma(const float* __restrict__ x,
                                                        float* __restrict__ out) {
    const int b    = blockIdx.x;        // batch row
    const int t    = threadIdx.x;       // 0..255
    const int wave = t >> 5;            // 0..7
    const int lane = t & 31;

    const float* xb  = x + b * ROW_LEN;
    const v4f*   xb4 = (const v4f*)xb;

    __shared__ float s_partial[8];

    // ---- Phase 1: per-wave sum of squares over 512 contiguous floats ----
    v8f c = {};
    const v2f ones = {1.0f, 1.0f};

    const int waveBase4 = wave * (512 / 4);   // float4 units
#pragma unroll
    for (int it = 0; it < 4; ++it) {
        v4f q = xb4[waveBase4 + it * 32 + lane];   // 512B coalesced per wave
        v2f a0 = {q.x * q.x, q.y * q.y};
        v2f a1 = {q.z * q.z, q.w * q.w};
        c = __builtin_amdgcn_wmma_f32_16x16x4_f32(false, a0, false, ones,
                                                  (short)0, c, false, false);
        c = __builtin_amdgcn_wmma_f32_16x16x4_f32(false, a1, false, ones,
                                                  (short)0, c, false, false);
    }

    // Half-column sum in this lane; pair with lane^16 for the full column.
    float colsum = c[0] + c[1] + c[2] + c[3] + c[4] + c[5] + c[6] + c[7];
    float s_wave = colsum + __shfl_xor(colsum, 16, 32);

    if (lane == 0) s_partial[wave] = s_wave;
    __syncthreads();

    float s = 0.0f;
#pragma unroll
    for (int w = 0; w < 8; ++w) s += s_partial[w];   // broadcast LDS reads
    const float scale = s * 0.0625f;                 // /16

    // ---- Phase 2: out[b,:] = x[b,:] * scale (float4 streaming) ----
    v4f* out4 = (v4f*)(out + b * ROW_LEN);
#pragma unroll
    for (int r = 0; r < 4; ++r) {
        v4f q = xb4[r * 256 + t];
        q *= scale;
        out4[r * 256 + t] = q;
    }
}

extern "C" void kernel_launch(void* const* d_in, const int* in_sizes, int n_in,
                              void* d_out, int out_size, void* d_ws, size_t ws_size,
                              hipStream_t stream) {
    (void)in_sizes; (void)n_in; (void)out_size; (void)d_ws; (void)ws_size;
    const float* x   = (const float*)d_in[0];
    float*       out = (float*)d_out;
    sumsq_scale_wmma<<<8, 256, 0, stream>>>(x, out);
}